// LSTM_74148315398336
// MI455X (gfx1250) — compile-verified
//
#include <hip/hip_runtime.h>
#include <hip/hip_bf16.h>

#define SEQ 262144
#define ISZ 60
#define HSZ 40
#define G4  160   // 4*HSZ

typedef float v2f __attribute__((ext_vector_type(2)));
typedef float v8f __attribute__((ext_vector_type(8)));

// ---------------------------------------------------------------------------
// Kernel 1: xg[s][g] = x[s][:] . w_ih[g][:] + b_ih[g] + b_hh[g]
// GEMM [S x 60] * [60 x 160] with V_WMMA_F32_16X16X4_F32 (full f32 precision).
// One wave per 16-row M-tile; 8 waves (256 threads) per block.
// ---------------------------------------------------------------------------
__global__ __launch_bounds__(256) void lstm_xg_wmma(
    const float* __restrict__ x, const float* __restrict__ w_ih,
    const float* __restrict__ b_ih, const float* __restrict__ b_hh,
    float* __restrict__ xg)
{
    const int lane  = threadIdx.x & 31;
    const int wave  = threadIdx.x >> 5;
    const int m0    = (blockIdx.x * 8 + wave) * 16;     // first row of this M-tile
    const int r16   = lane & 15;
    const int khalf = (lane >> 4) * 2;                  // 0 for lanes 0-15, 2 for 16-31
    const int hi8   = (lane >> 4) * 8;

    // A fragments: A[m][k], 16x4 per chunk, 15 chunks covering K=60.
    // Lane l<16 holds (K=0,1), lane>=16 holds (K=2,3) per ISA layout.
    v2f a[15];
    const float* xrow = x + (m0 + r16) * ISZ;
    #pragma unroll
    for (int kc = 0; kc < 15; ++kc) {
        a[kc].x = xrow[kc * 4 + khalf];
        a[kc].y = xrow[kc * 4 + khalf + 1];
    }

    // Loop over 10 N-tiles of 16 gate columns each.
    for (int nt = 0; nt < 10; ++nt) {
        const int col = nt * 16 + r16;                  // gate column this lane serves
        const float* wrow = w_ih + col * ISZ;           // B[k][col] = w_ih[col][k]
        v8f acc = {};
        #pragma unroll
        for (int kc = 0; kc < 15; ++kc) {
            v2f b;
            b.x = wrow[kc * 4 + khalf];
            b.y = wrow[kc * 4 + khalf + 1];
            acc = __builtin_amdgcn_wmma_f32_16x16x4_f32(
                false, a[kc], false, b, (short)0, acc, false, false);
        }
        const float bias = b_ih[col] + b_hh[col];
        #pragma unroll
        for (int r = 0; r < 8; ++r) {
            const int m = m0 + r + hi8;                 // VGPR r: rows r / r+8
            xg[m * G4 + col] = acc[r] + bias;
        }
    }
}

// ---------------------------------------------------------------------------
// Kernel 2: sequential LSTM scan, single workgroup of 160 threads (5 waves).
// Thread j owns gate row j: W_hh[j][0..39] held in 40 VGPRs.
// h lives in LDS (broadcast reads), c lives in registers of threads 0..39.
// xg (168 MB) fits in the 192 MB L2; prefetch 8 steps ahead.
// ---------------------------------------------------------------------------
__global__ __launch_bounds__(160) void lstm_scan(
    const float* __restrict__ xg, const float* __restrict__ w_hh,
    const float* __restrict__ w_lin, const float* __restrict__ b_lin,
    const float* __restrict__ h0, const float* __restrict__ c0,
    float* __restrict__ out, int S)
{
    __shared__ float sh_h[HSZ];
    __shared__ float sh_g[G4];

    const int j = threadIdx.x;

    float w[HSZ];
    #pragma unroll
    for (int k = 0; k < HSZ; ++k) w[k] = w_hh[j * HSZ + k];

    float cstate = 0.f;
    if (j < HSZ) { sh_h[j] = h0[j]; cstate = c0[j]; }
    __syncthreads();

    for (int t = 0; t < S; ++t) {
        float g = xg[t * G4 + j];
        if (t + 8 < S)
            __builtin_prefetch(&xg[(t + 8) * G4 + j], 0, 3);  // global_prefetch_b8

        // g += W_hh[j][:] . h  (LDS broadcast reads, b128 granularity)
        #pragma unroll
        for (int k0 = 0; k0 < HSZ; k0 += 4) {
            const float4 hv = *(const float4*)&sh_h[k0];
            g += w[k0] * hv.x + w[k0 + 1] * hv.y + w[k0 + 2] * hv.z + w[k0 + 3] * hv.w;
        }

        // Gate activation: rows [0,80) and [120,160) sigmoid; [80,120) tanh.
        float act;
        if (j < 80 || j >= 120) {
            act = 1.f / (1.f + __expf(-g));
        } else {
            const float gc = fminf(fmaxf(g, -15.f), 15.f);
            const float e  = __expf(-2.f * gc);
            act = (1.f - e) / (1.f + e);
        }
        sh_g[j] = act;
        __syncthreads();

        if (j < HSZ) {
            const float ig = sh_g[j];
            const float fg = sh_g[HSZ + j];
            const float gg = sh_g[2 * HSZ + j];
            const float og = sh_g[3 * HSZ + j];
            cstate = fg * cstate + ig * gg;
            const float cc = fminf(fmaxf(cstate, -15.f), 15.f);
            const float e  = __expf(-2.f * cc);
            sh_h[j] = og * (1.f - e) / (1.f + e);
        }
        __syncthreads();
    }

    if (j == 0) {
        float acc = b_lin[0];
        #pragma unroll
        for (int k = 0; k < HSZ; ++k) acc += w_lin[k] * sh_h[k];
        out[0] = acc;
    }
}

// ---------------------------------------------------------------------------
extern "C" void kernel_launch(void* const* d_in, const int* in_sizes, int n_in,
                              void* d_out, int out_size, void* d_ws, size_t ws_size,
                              hipStream_t stream) {
    const float* x     = (const float*)d_in[0];
    const float* w_ih  = (const float*)d_in[1];
    const float* w_hh  = (const float*)d_in[2];
    const float* b_ih  = (const float*)d_in[3];
    const float* b_hh  = (const float*)d_in[4];
    const float* w_lin = (const float*)d_in[5];
    const float* b_lin = (const float*)d_in[6];
    const float* h0    = (const float*)d_in[7];
    const float* c0    = (const float*)d_in[8];
    float* out = (float*)d_out;

    const int S = in_sizes[0] / ISZ;           // 262144
    float* xg = (float*)d_ws;                  // S * 160 floats = 168 MB scratch

    const int mtiles = S / 16;                 // 16384
    dim3 grid(mtiles / 8);                     // 8 waves (M-tiles) per block
    lstm_xg_wmma<<<grid, 256, 0, stream>>>(x, w_ih, b_ih, b_hh, xg);
    lstm_scan<<<1, 160, 0, stream>>>(xg, w_hh, w_lin, b_lin, h0, c0, out, S);
}